// SAGPool_16372415332891
// MI455X (gfx1250) — compile-verified
//
#include <hip/hip_runtime.h>
#include <math.h>

typedef float v2f __attribute__((ext_vector_type(2)));
typedef float v8f __attribute__((ext_vector_type(8)));

// ---------------------------------------------------------------------------
// Kernel 1: per-node dual dot products via V_WMMA_F32_16X16X4_F32.
//   s_rel[i] = x[i] . w_rel
//   base[i]  = x[i] . w_root + b_rel
//   agg[i]   = 0.0 (f64 accumulator init for edge scatter)
// One wave handles a 16-row tile; 16 chained K=4 f32 WMMAs cover K=64.
// B fragments are tile-invariant -> hoisted (branch-free cndmask selects).
// A loads are unconditional (row index clamped; garbage rows never stored).
// ---------------------------------------------------------------------------
__global__ void sag_node_dots(const float* __restrict__ x,
                              const float* __restrict__ w_rel,
                              const float* __restrict__ w_root,
                              const float* __restrict__ b_rel,
                              float* __restrict__ s_rel,
                              float* __restrict__ base,
                              double* __restrict__ agg,
                              int N) {
    const int lane = threadIdx.x & 31;
    const int l16  = lane & 15;      // M (A) / N (B) index
    const int kh   = lane >> 4;      // K-half select: 0 -> K{0,1}, 1 -> K{2,3}
    const int waveId = blockIdx.x * (blockDim.x >> 5) + (threadIdx.x >> 5);
    const int nWaves = gridDim.x * (blockDim.x >> 5);
    const int tiles  = (N + 15) >> 4;
    const float bias = b_rel[0];

    // ---- B fragments (tile-invariant): col0 = w_rel, col1 = w_root, rest 0.
    // Branch-free: pointer select + value mask, unconditional loads.
    const float* wsel = (l16 == 1) ? w_root : w_rel;   // v_cndmask on pointer
    const bool active = (l16 < 2);
    v2f barr[16];
#pragma unroll
    for (int kc = 0; kc < 16; ++kc) {
        const int kbase = (kc << 2) + (kh << 1);
        const float bx = wsel[kbase];
        const float by = wsel[kbase + 1];
        barr[kc].x = active ? bx : 0.0f;               // v_cndmask, no branch
        barr[kc].y = active ? by : 0.0f;
    }

    for (int tile = waveId; tile < tiles; tile += nWaves) {
        const int row0 = tile << 4;
        // clamp instead of predicating: OOB rows only affect C rows we never store
        int rowA = row0 + l16;
        if (rowA > N - 1) rowA = N - 1;                // v_min, no branch
        const float* __restrict__ xp = x + (size_t)rowA * 64 + (kh << 1);

        v8f c = {};
#pragma unroll
        for (int kc = 0; kc < 16; ++kc) {
            v2f a;
            a.x = xp[kc << 2];                         // merges to global_load_b64
            a.y = xp[(kc << 2) + 1];
            c = __builtin_amdgcn_wmma_f32_16x16x4_f32(
                    false, a, false, barr[kc], (short)0, c, false, false);
        }

        // C layout: VGPR r, lanes 0-15 -> M=r col=lane ; lanes 16-31 -> M=8+r
        if (l16 < 2) {
#pragma unroll
            for (int r = 0; r < 8; ++r) {
                const int row = row0 + (kh << 3) + r;
                if (row < N) {
                    if (l16 == 0) s_rel[row] = c[r];
                    else          base[row]  = c[r] + bias;
                }
            }
        }
        // zero the f64 edge accumulator for these rows
        if (kh == 0) {
            const int row = row0 + l16;
            if (row < N) agg[row] = 0.0;
        }
    }
}

// ---------------------------------------------------------------------------
// Kernel 2: edge scatter.  agg[dst] += s_rel[src]  (f64 atomics: order noise
// is ~2^-52 relative, vanishes when rounded to f32 -> replay-stable scores).
// s_rel (400KB) and agg (800KB) are L2-resident; 12.8MB edge stream dominates.
// ---------------------------------------------------------------------------
__global__ void sag_edge_scatter(const int* __restrict__ ei,
                                 const float* __restrict__ s_rel,
                                 double* __restrict__ agg,
                                 int E) {
    int e = blockIdx.x * blockDim.x + threadIdx.x;
    if (e < E) {
        const int src = ei[e];
        const int dst = ei[E + e];
        atomicAdd(&agg[dst], (double)s_rel[src]);
    }
}

// ---------------------------------------------------------------------------
// Kernel 3: per-graph top-k (rank via O(n^2) LDS compare; exact k kept since
// the comparator is a strict total order) + score-weighted mean pool.
// One 256-thread block per graph; graphs are contiguous (batch is sorted).
// ---------------------------------------------------------------------------
#define MAXG 8192   // >> max graph size (~500 for N=1e5, G=256)

__global__ void sag_topk_pool(const float* __restrict__ x,
                              const int* __restrict__ batch,
                              const float* __restrict__ base,
                              const double* __restrict__ agg,
                              float* __restrict__ out,
                              int N) {
    __shared__ float s_lds[MAXG];
    __shared__ float red[4 * 64];
    __shared__ int   seg[2];

    const int g = blockIdx.x;

    // graph boundaries: lower_bound(batch, g) and lower_bound(batch, g+1)
    if (threadIdx.x < 2) {
        const int target = g + (int)threadIdx.x;
        int lo = 0, hi = N;
        while (lo < hi) {
            const int mid = (lo + hi) >> 1;
            if (batch[mid] < target) lo = mid + 1; else hi = mid;
        }
        seg[threadIdx.x] = lo;
    }
    __syncthreads();
    const int start = seg[0];
    int n = seg[1] - seg[0];
    if (n > MAXG) n = MAXG;            // safety clamp (unreachable for harness)
    const int k = (n + 1) >> 1;        // ceil(0.5 * n)

    // scores -> LDS
    for (int i = threadIdx.x; i < n; i += blockDim.x)
        s_lds[i] = tanhf((float)(agg[start + i] + (double)base[start + i]));
    __syncthreads();

    // rank with stable index tie-break; keep iff rank < k
    unsigned keepmask = 0;              // up to MAXG/256 = 32 slots per thread
    for (int i = threadIdx.x, slot = 0; i < n; i += blockDim.x, ++slot) {
        const float si = s_lds[i];
        int rank = 0;
        for (int j = 0; j < n; ++j) {
            const float sj = s_lds[j];
            rank += (int)((sj > si) | ((sj == si) & (j < i)));
        }
        if (rank < k) keepmask |= (1u << slot);
    }
    __syncthreads();
    // overwrite scores with weights w_i = keep ? score : 0
    for (int i = threadIdx.x, slot = 0; i < n; i += blockDim.x, ++slot) {
        const float si = s_lds[i];
        s_lds[i] = ((keepmask >> slot) & 1u) ? si : 0.0f;
    }
    __syncthreads();

    // weighted pool: thread t handles dim d = t&63, node stride 4 (coalesced
    // 256B x-row reads); skip w==0 rows (exact: contribution is zero).
    const int d   = threadIdx.x & 63;
    const int grp = threadIdx.x >> 6;
    float acc = 0.0f;
    for (int i = grp; i < n; i += 4) {
        const float w = s_lds[i];
        if (w != 0.0f) acc += w * x[(size_t)(start + i) * 64 + d];
    }
    red[grp * 64 + d] = acc;
    __syncthreads();
    if (threadIdx.x < 64) {
        const float tot = red[d] + red[64 + d] + red[128 + d] + red[192 + d];
        out[(size_t)g * 64 + d] = tot / fmaxf((float)k, 1.0f);
    }
}

// ---------------------------------------------------------------------------
extern "C" void kernel_launch(void* const* d_in, const int* in_sizes, int n_in,
                              void* d_out, int out_size, void* d_ws, size_t ws_size,
                              hipStream_t stream) {
    const float* x      = (const float*)d_in[0];
    const int*   ei     = (const int*)  d_in[1];
    const int*   batch  = (const int*)  d_in[2];
    const float* w_rel  = (const float*)d_in[3];
    const float* b_rel  = (const float*)d_in[4];
    const float* w_root = (const float*)d_in[5];

    const int N = in_sizes[2];
    const int E = in_sizes[1] / 2;
    const int G = out_size / 64;
    float* out = (float*)d_out;

    // workspace layout: [agg f64 x N][s_rel f32 x N][base f32 x N] = 16N bytes
    char* ws = (char*)d_ws;
    double* agg   = (double*)ws;
    float*  s_rel = (float*)(ws + (size_t)N * 8);
    float*  base  = (float*)(ws + (size_t)N * 8 + (size_t)N * 4);

    const int tiles   = (N + 15) / 16;
    const int blocks1 = (tiles + 7) / 8;          // 8 waves per 256-thread block
    sag_node_dots<<<blocks1, 256, 0, stream>>>(x, w_rel, w_root, b_rel,
                                               s_rel, base, agg, N);
    sag_edge_scatter<<<(E + 255) / 256, 256, 0, stream>>>(ei, s_rel, agg, E);
    sag_topk_pool<<<G, 256, 0, stream>>>(x, batch, base, agg, out, N);
}